// LightGCNModel_8117488189798
// MI455X (gfx1250) — compile-verified
//
#include <hip/hip_runtime.h>

// Problem constants (match reference)
#define NUM_USERS 100000
#define NUM_ITEMS 50000
#define NNODES    (NUM_USERS + NUM_ITEMS)
#define DIM       64
#define NNZ_E     4000000
#define BATCH     8192
#define NUM_LAYERS 3

static const size_t ND = (size_t)NNODES * DIM;   // 9.6M floats = 38.4 MB

typedef float v2f __attribute__((ext_vector_type(2)));
typedef float v8f __attribute__((ext_vector_type(8)));

// ---------------------------------------------------------------------------
// E0 = concat(user_emb, item_emb); acc = E0
// ---------------------------------------------------------------------------
__global__ __launch_bounds__(256)
void lgcn_init(const float* __restrict__ uemb, const float* __restrict__ iemb,
               float* __restrict__ e, float* __restrict__ acc) {
  size_t i = (size_t)blockIdx.x * blockDim.x + threadIdx.x;
  const size_t stride = (size_t)gridDim.x * blockDim.x;
  const size_t nu = (size_t)NUM_USERS * DIM;
  for (; i < ND; i += stride) {
    float v = (i < nu) ? uemb[i] : iemb[i - nu];
    e[i] = v;
    acc[i] = v;
  }
}

// ---------------------------------------------------------------------------
// zero a float4 buffer
// ---------------------------------------------------------------------------
__global__ __launch_bounds__(256)
void lgcn_zero(float4* __restrict__ p, size_t n4) {
  size_t i = (size_t)blockIdx.x * blockDim.x + threadIdx.x;
  const size_t stride = (size_t)gridDim.x * blockDim.x;
  float4 z = make_float4(0.f, 0.f, 0.f, 0.f);
  for (; i < n4; i += stride) p[i] = z;
}

// ---------------------------------------------------------------------------
// COO SpMM: e_out[row] += val * e_in[col]   (one wave per edge; lane = 2 floats)
// Gather is a single coalesced 256B burst per edge; scatter is 64 f32 atomics
// into L2 (table fits in 192MB L2, so this is L2-atomic bound, not HBM bound).
// ---------------------------------------------------------------------------
__global__ __launch_bounds__(256)
void lgcn_spmm(const float* __restrict__ vals, const int* __restrict__ rows,
               const int* __restrict__ cols, const float* __restrict__ ein,
               float* __restrict__ eout) {
  const int lane = threadIdx.x & 31;
  const long long wid = (long long)blockIdx.x * (blockDim.x >> 5) + (threadIdx.x >> 5);
  const long long nw  = (long long)gridDim.x * (blockDim.x >> 5);
  for (long long e = wid; e < (long long)NNZ_E; e += nw) {
    // stream edge metadata one stride ahead (global_prefetch_b8)
    long long en = e + nw;
    if (en < (long long)NNZ_E) {
      __builtin_prefetch(&vals[en], 0, 3);
      __builtin_prefetch(&cols[en], 0, 3);
      __builtin_prefetch(&rows[en], 0, 3);
    }
    const float v = vals[e];
    const int r = rows[e];
    const int c = cols[e];
    const float2* __restrict__ src = (const float2*)(ein + (size_t)c * DIM);
    float2 x = src[lane];
    float* dst = eout + (size_t)r * DIM + (size_t)(lane * 2);
    atomicAdd(dst,     v * x.x);
    atomicAdd(dst + 1, v * x.y);
  }
}

// ---------------------------------------------------------------------------
// acc += e   (vectorized)
// ---------------------------------------------------------------------------
__global__ __launch_bounds__(256)
void lgcn_accum(float4* __restrict__ acc, const float4* __restrict__ e, size_t n4) {
  size_t i = (size_t)blockIdx.x * blockDim.x + threadIdx.x;
  const size_t stride = (size_t)gridDim.x * blockDim.x;
  for (; i < n4; i += stride) {
    float4 a = acc[i];
    float4 b = e[i];
    a.x += b.x; a.y += b.y; a.z += b.z; a.w += b.w;
    acc[i] = a;
  }
}

// ---------------------------------------------------------------------------
// BPR scoring with WMMA: one wave handles 16 triplets.
//   S = U(16x64) x (P-N)^T(64x16) via 16 chained V_WMMA_F32_16X16X4_F32;
//   out[b] = diag(S) / 16   (the two /4 layer-mean factors folded in).
// A frag (16x4 f32): lane l -> M = l&15; VGPR j, lane-half h -> K = j + 2h.
// B frag (4x16 f32): lane l -> N = l&15; VGPR j, lane-half h -> K = j + 2h.
// C/D (16x16 f32):  VGPR r: lanes 0-15 -> M=r, lanes 16-31 -> M=8+r.
// ---------------------------------------------------------------------------
__global__ __launch_bounds__(256)
void lgcn_score_wmma(const float* __restrict__ acc,
                     const int* __restrict__ uidx,
                     const int* __restrict__ pidx,
                     const int* __restrict__ nidx,
                     float* __restrict__ out) {
  const int lane = threadIdx.x & 31;
  const int wave = threadIdx.x >> 5;
  const int g = blockIdx.x * (blockDim.x >> 5) + wave;   // triplet group [0,512)
  const int mn = lane & 15;                              // M for A rows / N for B cols
  const int half = lane >> 4;                            // lane-half -> K pair select

  const int b0 = g * 16 + mn;
  const float* __restrict__ urow = acc + (size_t)uidx[b0] * DIM;
  const float* __restrict__ prow = acc + ((size_t)NUM_USERS + (size_t)pidx[b0]) * DIM;
  const float* __restrict__ nrow = acc + ((size_t)NUM_USERS + (size_t)nidx[b0]) * DIM;

  v8f c = {};
#pragma unroll
  for (int k = 0; k < 16; ++k) {
    const int kb = 4 * k + 2 * half;
    v2f a, b;
    a.x = urow[kb];
    a.y = urow[kb + 1];
    b.x = prow[kb]     - nrow[kb];
    b.y = prow[kb + 1] - nrow[kb + 1];
    // 8 args: (neg_a, A, neg_b, B, c_mod, C, reuse_a, reuse_b)
    c = __builtin_amdgcn_wmma_f32_16x16x4_f32(false, a, false, b,
                                              (short)0, c, false, false);
  }

  // Pull the diagonal out of the C/D layout:
  //  M=m (m<8):  VGPR m,   lane m        -> lanes 0..7
  //  M=m (m>=8): VGPR m-8, lane m+16     -> lanes 24..31
  const int idx = lane & 7;
  float dval = 0.f;
#pragma unroll
  for (int i = 0; i < 8; ++i)
    if (idx == i) dval = c[i];

  if (lane < 8) {
    out[g * 16 + lane] = dval * (1.0f / 16.0f);
  } else if (lane >= 24) {
    out[g * 16 + (lane - 16)] = dval * (1.0f / 16.0f);
  }
}

// ---------------------------------------------------------------------------
// Host launcher. ws layout: [acc | e_cur | e_next], 3 x 38.4MB = 115.2MB.
// ---------------------------------------------------------------------------
extern "C" void kernel_launch(void* const* d_in, const int* in_sizes, int n_in,
                              void* d_out, int out_size, void* d_ws, size_t ws_size,
                              hipStream_t stream) {
  (void)in_sizes; (void)n_in; (void)out_size; (void)ws_size;

  const float* user_emb = (const float*)d_in[0];
  const float* item_emb = (const float*)d_in[1];
  const float* adj_vals = (const float*)d_in[2];
  const int*   adj_rows = (const int*)d_in[3];
  const int*   adj_cols = (const int*)d_in[4];
  const int*   user_idx = (const int*)d_in[5];
  const int*   pos_idx  = (const int*)d_in[6];
  const int*   neg_idx  = (const int*)d_in[7];
  float* out = (float*)d_out;

  float* acc  = (float*)d_ws;
  float* e_a  = acc + ND;
  float* e_b  = e_a + ND;

  const size_t n4 = ND / 4;

  lgcn_init<<<4096, 256, 0, stream>>>(user_emb, item_emb, e_a, acc);

  for (int layer = 0; layer < NUM_LAYERS; ++layer) {
    lgcn_zero<<<2048, 256, 0, stream>>>((float4*)e_b, n4);
    lgcn_spmm<<<4096, 256, 0, stream>>>(adj_vals, adj_rows, adj_cols, e_a, e_b);
    lgcn_accum<<<2048, 256, 0, stream>>>((float4*)acc, (const float4*)e_b, n4);
    float* t = e_a; e_a = e_b; e_b = t;   // swap current/next
  }

  // 8192 triplets -> 512 wave-groups of 16 -> 64 blocks x 8 waves
  lgcn_score_wmma<<<BATCH / 16 / 8, 256, 0, stream>>>(acc, user_idx, pos_idx,
                                                      neg_idx, out);
}